// CEVP_54296976556743
// MI455X (gfx1250) — compile-verified
//
#include <hip/hip_runtime.h>
#include <hip/hip_bf16.h>
#include <stdint.h>

#define NUM_CLASSES 1784
#define BATCH       16384
#define ROW_F4      446      // 1784 / 4 float4 chunks per row (row = 7136 B, 16B aligned)
#define THREADS     256
#define NWAVES      8        // 256 threads / wave32

// One workgroup per row. Stage the row into LDS via CDNA5 async-to-LDS copies
// (ASYNCcnt path), pull each thread's 2 float4 chunks into registers once,
// then exact two-pass softmax (max+argmax, sum-exp) with wave32 reductions.
__global__ __launch_bounds__(THREADS) void cevp_row_kernel(
    const float* __restrict__ logits,
    const int*   __restrict__ targets,
    const float* __restrict__ penalty,
    float*       __restrict__ row_loss)
{
    __shared__ __align__(16) float srow[NUM_CLASSES];   // 7136 B staged row
    __shared__ float wred_v[NWAVES];
    __shared__ int   wred_i[NWAVES];
    __shared__ float wred_s[NWAVES];
    __shared__ float sM;
    __shared__ int   sAmax;

    const int row  = blockIdx.x;
    const int t    = threadIdx.x;
    const int lane = t & 31;
    const int wid  = t >> 5;

    const float* rowp = logits + (size_t)row * NUM_CLASSES;

    // ---- CDNA5 async global->LDS staging (ASYNCcnt, 16 B per lane per issue) ----
    // GVS mode: the immediate offset is added to both the global and the LDS address
    uint32_t lds_base = (uint32_t)(uintptr_t)(void*)&srow[0];
    uint32_t lofs = lds_base + (uint32_t)t * 16u;
    uint32_t mofs = (uint32_t)t * 16u;
    asm volatile("global_load_async_to_lds_b128 %0, %1, %2 offset:0"
                 :: "v"(lofs), "v"(mofs), "s"(rowp) : "memory");
    if (t < (ROW_F4 - THREADS)) {   // t < 190: second chunk, +4096 B on both addresses
        asm volatile("global_load_async_to_lds_b128 %0, %1, %2 offset:4096"
                     :: "v"(lofs), "v"(mofs), "s"(rowp) : "memory");
    }
    asm volatile("s_wait_asynccnt 0" ::: "memory");
    __syncthreads();

    // ---- load my 2 float4 chunks into registers (ds_load_b128, fixed trips) ----
    const float4* srow4 = (const float4*)srow;
    const float4 a = srow4[t];                       // cols 4t .. 4t+3
    const bool   hasB = (t < (ROW_F4 - THREADS));
    float4 b = make_float4(0.f, 0.f, 0.f, 0.f);
    if (hasB) b = srow4[t + THREADS];                // cols 4(t+256) .. +3

    // ---- pass 1: max + argmax (first occurrence on ties) ----
    float vmax = a.x;  int imax = 4 * t;
    if (a.y > vmax) { vmax = a.y; imax = 4 * t + 1; }
    if (a.z > vmax) { vmax = a.z; imax = 4 * t + 2; }
    if (a.w > vmax) { vmax = a.w; imax = 4 * t + 3; }
    if (hasB) {
        const int bb = 4 * (t + THREADS);
        if (b.x > vmax) { vmax = b.x; imax = bb;     }
        if (b.y > vmax) { vmax = b.y; imax = bb + 1; }
        if (b.z > vmax) { vmax = b.z; imax = bb + 2; }
        if (b.w > vmax) { vmax = b.w; imax = bb + 3; }
    }
    for (int m = 16; m > 0; m >>= 1) {               // wave32 tree
        float ov = __shfl_xor(vmax, m, 32);
        int   oi = __shfl_xor(imax, m, 32);
        if (ov > vmax || (ov == vmax && oi < imax)) { vmax = ov; imax = oi; }
    }
    if (lane == 0) { wred_v[wid] = vmax; wred_i[wid] = imax; }
    __syncthreads();
    if (t == 0) {
        float bv = wred_v[0]; int bi = wred_i[0];
        for (int w = 1; w < NWAVES; ++w) {
            float v = wred_v[w]; int i = wred_i[w];
            if (v > bv || (v == bv && i < bi)) { bv = v; bi = i; }
        }
        sM = bv; sAmax = bi;
    }
    __syncthreads();
    const float M = sM;

    // ---- pass 2: sum of exp(x - M) from registers ----
    float s = expf(a.x - M) + expf(a.y - M) + expf(a.z - M) + expf(a.w - M);
    if (hasB)
        s += expf(b.x - M) + expf(b.y - M) + expf(b.z - M) + expf(b.w - M);
    for (int m = 16; m > 0; m >>= 1) s += __shfl_xor(s, m, 32);
    if (lane == 0) wred_s[wid] = s;
    __syncthreads();

    if (t == 0) {
        float S = 0.f;
        for (int w = 0; w < NWAVES; ++w) S += wred_s[w];
        const int tgt = targets[row];
        const float ce  = -(srow[tgt] - M - logf(S));
        const float pen = penalty[(size_t)tgt * NUM_CLASSES + sAmax];
        row_loss[row] = ce + pen;   // W_PENALTY == 1
    }
}

// Deterministic final mean over the 16384 per-row partials (fixed-order tree,
// no fp atomics -> bitwise-stable across graph replays).
__global__ __launch_bounds__(256) void cevp_reduce_kernel(
    const float* __restrict__ row_loss, float* __restrict__ out)
{
    __shared__ float part[256];
    const int t = threadIdx.x;
    float s = 0.f;
    for (int i = t; i < BATCH; i += 256) s += row_loss[i];
    part[t] = s;
    __syncthreads();
    for (int off = 128; off > 0; off >>= 1) {
        if (t < off) part[t] += part[t + off];
        __syncthreads();
    }
    if (t == 0) out[0] = part[0] * (1.0f / (float)BATCH);
}

extern "C" void kernel_launch(void* const* d_in, const int* in_sizes, int n_in,
                              void* d_out, int out_size, void* d_ws, size_t ws_size,
                              hipStream_t stream)
{
    (void)in_sizes; (void)n_in; (void)out_size; (void)ws_size;
    const float* logits  = (const float*)d_in[0];
    const int*   targets = (const int*)d_in[1];
    const float* penalty = (const float*)d_in[2];
    float* row_loss = (float*)d_ws;          // 16384 floats of scratch

    cevp_row_kernel<<<BATCH, THREADS, 0, stream>>>(logits, targets, penalty, row_loss);
    cevp_reduce_kernel<<<1, 256, 0, stream>>>(row_loss, (float*)d_out);
}